// local_ssl_46875273069282
// MI455X (gfx1250) — compile-verified
//
#include <hip/hip_runtime.h>

typedef float v2f __attribute__((ext_vector_type(2)));
typedef float v8f __attribute__((ext_vector_type(8)));

#define BATCH   4096
#define TWOB    8192
#define EMBED   64
#define SQRT_INV_T 2.2360679775f   // sqrt(1/0.2); folded into Q on both sides
#define LDS_STR 68                 // padded row stride (floats) -> no bank conflicts

// ---------------------------------------------------------------------------
// Kernel 1: gather (+ L2-normalize for user branch) into Q, pre-scaled by
// sqrt(1/T) so the Gram product directly yields logits (score/T).
// one 64-thread block per output row; rows [0,8192) = user, [8192,16384) = item
// ---------------------------------------------------------------------------
__global__ __launch_bounds__(64) void prep_kernel(
    const int* __restrict__ u_idx, const int* __restrict__ i_idx,
    const float* __restrict__ u1e, const float* __restrict__ i1e,
    const float* __restrict__ u2e, const float* __restrict__ i2e,
    float* __restrict__ qu, float* __restrict__ qi)
{
    __shared__ float red[64];
    int row  = blockIdx.x;           // 0..16383
    int t    = threadIdx.x;          // 0..63
    bool user = row < TWOB;
    int r    = user ? row : row - TWOB;
    int half = (r >= BATCH) ? 1 : 0; // 0 -> sub1 view, 1 -> sub2 view
    int b    = half ? r - BATCH : r;
    const int*   idx = user ? u_idx : i_idx;
    const float* src = user ? (half ? u2e : u1e) : (half ? i2e : i1e);
    int e = idx[b];
    float v = src[(size_t)e * EMBED + t];
    float scale = SQRT_INV_T;
    if (user) {
        red[t] = v * v;
        __syncthreads();
        for (int off = 32; off > 0; off >>= 1) {
            if (t < off) red[t] += red[t + off];
            __syncthreads();
        }
        scale = SQRT_INV_T / sqrtf(red[0]);
    }
    float* dst = user ? qu : qi;
    dst[(size_t)r * EMBED + t] = v * scale;
}

// ---------------------------------------------------------------------------
// Kernel 2: streaming Gram(Q) + online logsumexp, fp32 WMMA 16x16x4.
// grid = (TWOB/128, 2 branches), block = 256 threads (8 waves, 16 rows each).
// Register-prefetch pipeline hides the global->LDS round trip; elementwise
// online-softmax update is fully branchless (cndmask, no exec churn).
// ---------------------------------------------------------------------------
__global__ __launch_bounds__(256) void gram_lse_kernel(
    const float* __restrict__ qu, const float* __restrict__ qi,
    float* __restrict__ row_loss)
{
    __shared__ float kt[16 * LDS_STR];

    const float* Q = (blockIdx.y == 0) ? qu : qi;
    int tid  = threadIdx.x;
    int wave = tid >> 5;
    int lane = tid & 31;
    int ln   = lane & 15;           // N (column within tile) / M (row within strip)
    int hi   = lane >> 4;           // selects K sub-pair {2,3} vs {0,1}
    int strip = blockIdx.x * 128 + wave * 16;
    int arow  = strip + ln;

    // A operands: fixed for the whole column sweep (16 x v2f = 32 VGPRs)
    v2f a[16];
    {
        const float* qr = Q + (size_t)arow * EMBED;
#pragma unroll
        for (int k = 0; k < 16; ++k) {
            int o = 4 * k + 2 * hi;
            a[k].x = qr[o];
            a[k].y = qr[o + 1];
        }
    }

    // per-C-VGPR-row constants, hoisted out of the sweep
    int rowv[8], ptgt[8];
#pragma unroll
    for (int r = 0; r < 8; ++r) {
        rowv[r] = strip + r + 8 * hi;
        ptgt[r] = rowv[r] ^ BATCH;   // positive column: (row + B) mod 2B
    }

    // lane-local running (max, sumexp) per row, plus positive-logit capture
    float m[8], s[8], pv[8];
#pragma unroll
    for (int r = 0; r < 8; ++r) { m[r] = -3.0e38f; s[r] = 0.0f; pv[r] = 0.0f; }

    // cooperative-load coordinates (256 thr x float4 = 16x64 tile)
    int ldr = tid >> 4;              // 0..15
    int ldc = (tid & 15) * 4;        // 0..60
    float4 vv = *(const float4*)(Q + (size_t)ldr * EMBED + ldc);  // tile 0

    for (int t = 0; t < TWOB / 16; ++t) {
        __syncthreads();             // all waves done reading kt
        {
            float* dd = &kt[ldr * LDS_STR + ldc];
            dd[0] = vv.x; dd[1] = vv.y; dd[2] = vv.z; dd[3] = vv.w;
        }
        if (t + 1 < TWOB / 16)       // prefetch next tile; latency hidden by compute
            vv = *(const float4*)(Q + (size_t)((t + 1) * 16 + ldr) * EMBED + ldc);
        __syncthreads();             // kt ready

        v8f c = {0.f, 0.f, 0.f, 0.f, 0.f, 0.f, 0.f, 0.f};
#pragma unroll
        for (int k = 0; k < 16; ++k) {
            v2f b;
            const float* kr = &kt[ln * LDS_STR + 4 * k + 2 * hi];
            b.x = kr[0];
            b.y = kr[1];
            c = __builtin_amdgcn_wmma_f32_16x16x4_f32(
                    false, a[k], false, b, (short)0, c, false, false);
        }

        int col = t * 16 + ln;
#pragma unroll
        for (int r = 0; r < 8; ++r) {
            float raw  = c[r];                     // already in logit units
            bool  diag = (col == rowv[r]);
            float nm   = diag ? m[r] : fmaxf(m[r], raw);
            float e2   = diag ? 0.0f : __expf(raw - nm);  // +inf selected away
            s[r] = fmaf(s[r], __expf(m[r] - nm), e2);
            m[r] = nm;
            pv[r] += (col == ptgt[r]) ? raw : 0.0f;
        }
    }

    // merge the 16 lane-partial (m,s) pairs per row; sum positive captures
#pragma unroll
    for (int r = 0; r < 8; ++r) {
        for (int off = 8; off >= 1; off >>= 1) {
            float om = __shfl_xor(m[r],  off, 16);
            float os = __shfl_xor(s[r],  off, 16);
            float op = __shfl_xor(pv[r], off, 16);
            float nm = fmaxf(m[r], om);
            s[r]  = fmaf(s[r], __expf(m[r] - nm), os * __expf(om - nm));
            m[r]  = nm;
            pv[r] += op;
        }
    }
    if (ln == 0) {
        // lane 0 (hi=0): rows strip..strip+7 ; lane 16 (hi=1): rows strip+8..+15
#pragma unroll
        for (int r = 0; r < 8; ++r) {
            row_loss[(size_t)blockIdx.y * TWOB + rowv[r]] =
                (m[r] + __logf(s[r])) - pv[r];
        }
    }
}

// ---------------------------------------------------------------------------
// Kernel 3: deterministic fixed-order reduction of 16384 row losses -> scalar
// ---------------------------------------------------------------------------
__global__ __launch_bounds__(256) void reduce_kernel(
    const float* __restrict__ row_loss, float* __restrict__ out)
{
    __shared__ float red[256];
    int t = threadIdx.x;
    float acc = 0.0f;
    for (int i = t; i < 2 * TWOB; i += 256) acc += row_loss[i];
    red[t] = acc;
    __syncthreads();
    for (int off = 128; off > 0; off >>= 1) {
        if (t < off) red[t] += red[t + off];
        __syncthreads();
    }
    if (t == 0) out[0] = red[0] * (1.0f / (float)TWOB);
}

// ---------------------------------------------------------------------------
extern "C" void kernel_launch(void* const* d_in, const int* in_sizes, int n_in,
                              void* d_out, int out_size, void* d_ws, size_t ws_size,
                              hipStream_t stream) {
    const int*   u_idx = (const int*)d_in[0];
    const int*   i_idx = (const int*)d_in[1];
    const float* u1e   = (const float*)d_in[2];
    const float* i1e   = (const float*)d_in[3];
    const float* u2e   = (const float*)d_in[4];
    const float* i2e   = (const float*)d_in[5];

    float* qu = (float*)d_ws;                       // [8192, 64]
    float* qi = qu + (size_t)TWOB * EMBED;          // [8192, 64]
    float* rl = qi + (size_t)TWOB * EMBED;          // [16384]

    prep_kernel<<<2 * TWOB, 64, 0, stream>>>(u_idx, i_idx, u1e, i1e, u2e, i2e, qu, qi);
    gram_lse_kernel<<<dim3(TWOB / 128, 2), 256, 0, stream>>>(qu, qi, rl);
    reduce_kernel<<<1, 256, 0, stream>>>(rl, (float*)d_out);
}